// Attention_33492154974379
// MI455X (gfx1250) — compile-verified
//
#include <hip/hip_runtime.h>
#include <hip/hip_bf16.h>

typedef float v2f __attribute__((ext_vector_type(2)));
typedef float v8f __attribute__((ext_vector_type(8)));

#define ATT_B 8
#define ATT_T 2048
#define ATT_D 1024
#define NEG_INF -1.0e9f

#if defined(__HIP_DEVICE_COMPILE__) && __has_builtin(__builtin_amdgcn_wmma_f32_16x16x4_f32)
#define HAVE_WMMA_F32X4 1
static __device__ __forceinline__ v8f wmma_f32(v2f a, v2f b, v8f c) {
  // D = A(16x4, f32) * B(4x16, f32) + C(16x16, f32)
  return __builtin_amdgcn_wmma_f32_16x16x4_f32(
      /*neg_a=*/false, a, /*neg_b=*/false, b,
      /*c_mod=*/(short)0, c, /*reuse_a=*/false, /*reuse_b=*/false);
}
#else
#define HAVE_WMMA_F32X4 0
// Canary fallback so the TU always compiles; histogram will show wmma=0.
static __device__ __forceinline__ v8f wmma_f32(v2f a, v2f b, v8f c) {
  c[0] += a[0] * b[0];
  return c;
}
#endif

// ---------------------------------------------------------------------------
// Kernel 1: scores = mask(x x^T), row softmax, write normalized weights.
// Grid: B * (T/16) workgroups, 512 threads (16 waves). Each WG owns 16 rows.
// Wave w owns column tiles c0 = w*128 + j*16, j in [0,8).
// WMMA f32 16x16x4 fragment mapping (ISA 7.12.2):
//   A: lane L -> M = L&15 ; VGPR{0,1} = K{2h, 2h+1},  h = L>>4
//   B: lane L -> N = L&15 ; VGPR{0,1} = K{2h, 2h+1}
//   C: lane L, VGPR v -> M = v + 8h, N = L&15
// ---------------------------------------------------------------------------
__global__ __launch_bounds__(512) void attn_scores_softmax_kernel(
    const float* __restrict__ x, float* __restrict__ w_out) {
  const int T = ATT_T, D = ATT_D;
  const int wg = blockIdx.x;
  const int b  = wg >> 7;          // T/16 = 128 row tiles per batch
  const int rt = wg & 127;
  const int r0 = rt << 4;

  const int tid  = threadIdx.x;
  const int wave = tid >> 5;       // 0..15
  const int lane = tid & 31;
  const int lm   = lane & 15;
  const int h    = lane >> 4;

  const float* __restrict__ xb = x + (size_t)b * T * D;
  // A fragment row pointer: x[b, r0+lm, 2h + k0]
  const float* aPtr = xb + (size_t)(r0 + lm) * D + 2 * h;
  // B fragment row pointer base: x[b, wave*128 + lm, 2h + k0] (+ j*16 rows)
  const float* bPtr = xb + (size_t)(wave * 128 + lm) * D + 2 * h;

  __builtin_prefetch(aPtr, 0, 0);
  __builtin_prefetch(bPtr, 0, 0);

  v8f acc[8] = {};

  for (int k0 = 0; k0 < D; k0 += 4) {
    const v2f a = *(const v2f*)(aPtr + k0);
#pragma unroll
    for (int j = 0; j < 8; ++j) {
      const v2f bb = *(const v2f*)(bPtr + k0 + j * 16 * D);
      acc[j] = wmma_f32(a, bb, acc[j]);
    }
  }

  // Causal mask: keep strictly-lower (s < t), else -1e9 (row 0 -> all -1e9).
#pragma unroll
  for (int j = 0; j < 8; ++j) {
    const int s = wave * 128 + j * 16 + lm;
#pragma unroll
    for (int v = 0; v < 8; ++v) {
      const int t = r0 + v + 8 * h;
      acc[j][v] = (s < t) ? acc[j][v] : NEG_INF;
    }
  }

  // Row max: per-register across 8 tiles, then across the 16-lane half.
  float rmax[8];
#pragma unroll
  for (int v = 0; v < 8; ++v) {
    float m = acc[0][v];
#pragma unroll
    for (int j = 1; j < 8; ++j) m = fmaxf(m, acc[j][v]);
    rmax[v] = m;
  }
#pragma unroll
  for (int mask = 1; mask <= 8; mask <<= 1) {
#pragma unroll
    for (int v = 0; v < 8; ++v)
      rmax[v] = fmaxf(rmax[v], __shfl_xor(rmax[v], mask, 32));
  }

  __shared__ float red[16][16];  // [wave][row-in-tile]
  if (lm == 0) {
#pragma unroll
    for (int v = 0; v < 8; ++v) red[wave][v + 8 * h] = rmax[v];
  }
  __syncthreads();

  float gmax[8];
#pragma unroll
  for (int v = 0; v < 8; ++v) {
    float m = red[0][v + 8 * h];
#pragma unroll
    for (int w2 = 1; w2 < 16; ++w2) m = fmaxf(m, red[w2][v + 8 * h]);
    gmax[v] = m;
  }
  __syncthreads();

  // exp(score - max) and row sums.
  float rsum[8];
#pragma unroll
  for (int v = 0; v < 8; ++v) rsum[v] = 0.0f;
#pragma unroll
  for (int j = 0; j < 8; ++j) {
#pragma unroll
    for (int v = 0; v < 8; ++v) {
      const float e = __expf(acc[j][v] - gmax[v]);
      acc[j][v] = e;
      rsum[v] += e;
    }
  }
#pragma unroll
  for (int mask = 1; mask <= 8; mask <<= 1) {
#pragma unroll
    for (int v = 0; v < 8; ++v) rsum[v] += __shfl_xor(rsum[v], mask, 32);
  }
  if (lm == 0) {
#pragma unroll
    for (int v = 0; v < 8; ++v) red[wave][v + 8 * h] = rsum[v];
  }
  __syncthreads();

  float inv[8];
#pragma unroll
  for (int v = 0; v < 8; ++v) {
    float s = 0.0f;
#pragma unroll
    for (int w2 = 0; w2 < 16; ++w2) s += red[w2][v + 8 * h];
    inv[v] = 1.0f / s;
  }

  // Store normalized weights: w[b, t, s].
  float* __restrict__ wb = w_out + (size_t)b * T * T;
#pragma unroll
  for (int j = 0; j < 8; ++j) {
    const int s = wave * 128 + j * 16 + lm;
#pragma unroll
    for (int v = 0; v < 8; ++v) {
      const int t = r0 + v + 8 * h;
      wb[(size_t)t * T + s] = acc[j][v] * inv[v];
    }
  }
}

// ---------------------------------------------------------------------------
// Kernel 2: att_vec = W @ x, exploiting causality (col tiles <= row tile;
// row-tile 0 uses the full width so the uniform row 0 is correct).
// Grid: B * (T/16) workgroups, 512 threads. Wave w owns N-tiles
// d0 = w*64 + j*16, j in [0,4) -> covers D = 1024.
// ---------------------------------------------------------------------------
__global__ __launch_bounds__(512) void attn_av_kernel(
    const float* __restrict__ x, const float* __restrict__ w_in,
    float* __restrict__ out) {
  const int T = ATT_T, D = ATT_D;
  const int wg = blockIdx.x;
  const int b  = wg >> 7;
  const int rt = wg & 127;
  const int r0 = rt << 4;

  const int tid  = threadIdx.x;
  const int wave = tid >> 5;
  const int lane = tid & 31;
  const int lm   = lane & 15;
  const int h    = lane >> 4;

  const float* __restrict__ xb = x + (size_t)b * T * D;
  const float* __restrict__ wb = w_in + (size_t)b * T * T;

  // A fragment: W[b, r0+lm, 2h + k0 .. +1] (contiguous float2 in a row).
  const float* aPtr = wb + (size_t)(r0 + lm) * T + 2 * h;
  // B fragment: x[b, k0+2h (+1), d0+lm] -> two strided b32 loads.
  const float* bBase = xb + (size_t)(2 * h) * D + lm;

  const int kmax = (rt == 0) ? T : ((rt + 1) << 4);

  v8f acc[4] = {};

  for (int k0 = 0; k0 < kmax; k0 += 4) {
    const v2f a = *(const v2f*)(aPtr + k0);
#pragma unroll
    for (int j = 0; j < 4; ++j) {
      const int d0 = wave * 64 + j * 16;
      v2f bb;
      bb.x = bBase[(size_t)k0 * D + d0];
      bb.y = bBase[(size_t)k0 * D + D + d0];
      acc[j] = wmma_f32(a, bb, acc[j]);
    }
  }

  float* __restrict__ ob = out + (size_t)b * T * D;
#pragma unroll
  for (int j = 0; j < 4; ++j) {
    const int d0 = wave * 64 + j * 16;
#pragma unroll
    for (int v = 0; v < 8; ++v) {
      const int t = r0 + v + 8 * h;
      ob[(size_t)t * D + d0 + lm] = acc[j][v];
    }
  }
}

// ---------------------------------------------------------------------------
extern "C" void kernel_launch(void* const* d_in, const int* in_sizes, int n_in,
                              void* d_out, int out_size, void* d_ws, size_t ws_size,
                              hipStream_t stream) {
  (void)in_sizes; (void)n_in; (void)out_size; (void)d_ws; (void)ws_size;
  const float* x = (const float*)d_in[0];
  float* out = (float*)d_out;

  // d_out layout (return order): att_vec [B,T,D] then att_weights [B,T,T].
  float* att_vec = out;
  float* att_w   = out + (size_t)ATT_B * ATT_T * ATT_D;

  const int n_wg = ATT_B * (ATT_T / 16);  // 1024
  attn_scores_softmax_kernel<<<n_wg, 512, 0, stream>>>(x, att_w);
  attn_av_kernel<<<n_wg, 512, 0, stream>>>(x, att_w, att_vec);
}